// BPR__71116068487556
// MI455X (gfx1250) — compile-verified
//
#include <hip/hip_runtime.h>
#include <hip/hip_bf16.h>

#define FACTOR 64

typedef float v2f __attribute__((ext_vector_type(2)));
typedef float v8f __attribute__((ext_vector_type(8)));

// Native f32 atomic add (no return) at device scope: guarantees a single
// global_atomic_add_f32 instruction (L2 RMW) instead of a CAS retry loop.
// volatile (no "memory" clobber) keeps it emitted & ordered vs other atomics
// while still allowing independent gather loads to be scheduled around it.
// STOREcnt from the asm is untracked by the compiler; s_endpgm's implicit
// wait-idle covers it and nothing in-kernel reads y.
__device__ __forceinline__ void atomic_add_f32_dev(float* p, float v) {
  unsigned long long a = (unsigned long long)p;
  asm volatile("global_atomic_add_f32 %0, %1, off scope:SCOPE_DEV"
               :: "v"(a), "v"(v));
}

// ---------------------------------------------------------------------------
// COO SpMM: y[rows[e], :] += vals[e] * x[cols[e], :]
// One wave32 per 4 consecutive edges; each lane owns 2 features (float2).
// Edge data is wave-uniform -> readfirstlane + int4/float4 loads so the
// compiler can use the scalar (s_load_b128 / KMcnt) path. Gathers hit the
// 192MB L2 (tables are 12-15MB); scatter = device-scope global_atomic_add_f32.
// Four independent gather->atomic chains per wave hide L2 latency.
// ---------------------------------------------------------------------------
__global__ __launch_bounds__(256) void spmm_coo_atomic(
    const int* __restrict__ rows, const int* __restrict__ cols,
    const float* __restrict__ vals, const float* __restrict__ x,
    float* __restrict__ y, int nnz) {
  const int lane = threadIdx.x;                              // 0..31
  const int waveId = blockIdx.x * blockDim.y + threadIdx.y;  // uniform per wave
  const int e0 = waveId * 4;
  if (e0 >= nnz) return;

  // Stay ~16KB ahead of the linear edge stream (gfx1250 global_prefetch_b8)
  __builtin_prefetch(rows + e0 + 4096, 0, 1);
  __builtin_prefetch(cols + e0 + 4096, 0, 1);
  __builtin_prefetch(vals + e0 + 4096, 0, 1);

  if (e0 + 3 < nnz) {
    const int eu = __builtin_amdgcn_readfirstlane(e0);       // force SGPR index
    const int4   r4 = *(const int4*)(rows + eu);
    const int4   c4 = *(const int4*)(cols + eu);
    const float4 v4 = *(const float4*)(vals + eu);
    const int   rr[4] = {r4.x, r4.y, r4.z, r4.w};
    const int   cc[4] = {c4.x, c4.y, c4.z, c4.w};
    const float vv[4] = {v4.x, v4.y, v4.z, v4.w};
#pragma unroll
    for (int k = 0; k < 4; ++k) {
      const float2 xv = ((const float2*)(x + (size_t)cc[k] * FACTOR))[lane];
      float* yp = y + (size_t)rr[k] * FACTOR + (size_t)(lane * 2);
      atomic_add_f32_dev(yp + 0, xv.x * vv[k]);
      atomic_add_f32_dev(yp + 1, xv.y * vv[k]);
    }
  } else {
    for (int k = 0; k < 4; ++k) {
      const int e = e0 + k;
      if (e >= nnz) break;
      const int r = rows[e];
      const int c = cols[e];
      const float v = vals[e];
      const float2 xv = ((const float2*)(x + (size_t)c * FACTOR))[lane];
      float* yp = y + (size_t)r * FACTOR + (size_t)(lane * 2);
      atomic_add_f32_dev(yp + 0, xv.x * v);
      atomic_add_f32_dev(yp + 1, xv.y * v);
    }
  }
}

// ---------------------------------------------------------------------------
// Epilogue via V_WMMA_F32_16X16X4_F32:
//   out[r, n] = 1*x0 + 0.5*x1 + (1/3)*x2 + 0.25*x3      (K=4 contraction)
// A (16x4): coefficients broadcast down M (lanes<16: K0,K1; lanes>=16: K2,K3).
// B (4x16): one row r, 16 features; lane half selects which hop pair it
// carries -> done branchlessly by selecting base POINTERS per lane, so loads
// and stores run with full EXEC (no saveexec churn around the WMMA).
// D rows are identical over M: every lane stores c[0]; lanes 16-31 write the
// same value to the same address as lanes 0-15 (benign duplicate).
// ---------------------------------------------------------------------------
__global__ __launch_bounds__(128) void combine_wmma(
    const float* __restrict__ x0, const float* __restrict__ x1,
    const float* __restrict__ x2, const float* __restrict__ x3,
    float* __restrict__ out, int nrows) {
  const int lane = threadIdx.x & 31;
  const int wave = threadIdx.x >> 5;
  const int row = blockIdx.x * (blockDim.x >> 5) + wave;
  if (row >= nrows) return;   // uniform per wave -> EXEC all-1s for WMMA

  const bool lo = (lane < 16);
  v2f a;
  a.x = lo ? 1.0f : (1.0f / 3.0f);
  a.y = lo ? 0.5f : 0.25f;

  // Branchless source selection: pointer cndmask instead of EXEC divergence.
  const float* __restrict__ p0 = lo ? x0 : x2;   // B VGPR0 source (K0 / K2)
  const float* __restrict__ p1 = lo ? x1 : x3;   // B VGPR1 source (K1 / K3)

  const size_t base = (size_t)row * FACTOR;
  const int n = lane & 15;

#pragma unroll
  for (int nb = 0; nb < FACTOR; nb += 16) {
    v2f b;
    b.x = p0[base + nb + n];
    b.y = p1[base + nb + n];
    v8f c = {};
    c = __builtin_amdgcn_wmma_f32_16x16x4_f32(
        /*neg_a=*/false, a, /*neg_b=*/false, b,
        /*c_mod=*/(short)0, c, /*reuse_a=*/false, /*reuse_b=*/false);
    out[base + nb + n] = c[0];
  }
}

// ---------------------------------------------------------------------------
// Launch: 6 dependent SpMMs (stream order enforces hop dependencies),
// then WMMA epilogues for users and items.
// ---------------------------------------------------------------------------
extern "C" void kernel_launch(void* const* d_in, const int* in_sizes, int n_in,
                              void* d_out, int out_size, void* d_ws, size_t ws_size,
                              hipStream_t stream) {
  (void)n_in; (void)out_size; (void)ws_size;

  const float* embed_user = (const float*)d_in[0];
  const float* embed_item = (const float*)d_in[1];
  const int*   u_idx      = (const int*)d_in[2];
  const int*   i_idx      = (const int*)d_in[3];
  const float* vals_ui    = (const float*)d_in[4];
  const float* vals_iu    = (const float*)d_in[5];

  const int U   = in_sizes[0] / FACTOR;
  const int I   = in_sizes[1] / FACTOR;
  const int nnz = in_sizes[2];

  // Workspace: six accumulators g{1,2,3}{u,i}  (~80 MB total)
  float* g1u = (float*)d_ws;
  float* g1i = g1u + (size_t)U * FACTOR;
  float* g2u = g1i + (size_t)I * FACTOR;
  float* g2i = g2u + (size_t)U * FACTOR;
  float* g3u = g2i + (size_t)I * FACTOR;
  float* g3i = g3u + (size_t)U * FACTOR;

  const size_t zbytes = (size_t)3 * (size_t)(U + I) * FACTOR * sizeof(float);
  hipMemsetAsync(d_ws, 0, zbytes, stream);

  dim3 sblk(32, 8, 1);                            // 8 waves/block, 4 edges/wave
  const int waves = (nnz + 3) / 4;
  const int sgrid = (waves + 7) / 8;

  // hop 1
  spmm_coo_atomic<<<sgrid, sblk, 0, stream>>>(u_idx, i_idx, vals_ui, embed_item, g1u, nnz);
  spmm_coo_atomic<<<sgrid, sblk, 0, stream>>>(i_idx, u_idx, vals_iu, embed_user, g1i, nnz);
  // hop 2
  spmm_coo_atomic<<<sgrid, sblk, 0, stream>>>(u_idx, i_idx, vals_ui, g1i, g2u, nnz);
  spmm_coo_atomic<<<sgrid, sblk, 0, stream>>>(i_idx, u_idx, vals_iu, g1u, g2i, nnz);
  // hop 3
  spmm_coo_atomic<<<sgrid, sblk, 0, stream>>>(u_idx, i_idx, vals_ui, g2i, g3u, nnz);
  spmm_coo_atomic<<<sgrid, sblk, 0, stream>>>(i_idx, u_idx, vals_iu, g2u, g3i, nnz);

  // epilogue (WMMA f32 16x16x4)
  float* out_users = (float*)d_out;
  float* out_items = out_users + (size_t)U * FACTOR;
  const int wpb = 4;                              // waves per block
  combine_wmma<<<(U + wpb - 1) / wpb, wpb * 32, 0, stream>>>(embed_user, g1u, g2u, g3u, out_users, U);
  combine_wmma<<<(I + wpb - 1) / wpb, wpb * 32, 0, stream>>>(embed_item, g1i, g2i, g3i, out_items, I);
}